// NURBSSurface_20066087207483
// MI455X (gfx1250) — compile-verified
//
#include <hip/hip_runtime.h>
#include <hip/hip_bf16.h>

#define NCP 32
#define NTHREADS 256

// knots = [0,0,0, linspace(0,1,30), 1,1,1], i.e. knot(i) = clamp((i-3)/29, 0, 1)
__device__ __forceinline__ float knotv(int i) {
    float t = (float)(i - 3) * (1.0f / 29.0f);
    return fminf(fmaxf(t, 0.0f), 1.0f);
}

// Cubic basis functions + first derivatives (exact transcription of the
// reference Cox-de Boor ndu recursion for p=3, n=1; divisions -> v_rcp_f32).
__device__ __forceinline__ void basis3(float u, int span, float* N, float* D) {
    float left[4], right[4];
    float ndu[4][4];
    ndu[0][0] = 1.0f;
#pragma unroll
    for (int j = 1; j <= 3; ++j) {
        left[j]  = u - knotv(span + 1 - j);
        right[j] = knotv(span + j) - u;
        float saved = 0.0f;
#pragma unroll
        for (int r = 0; r < j; ++r) {
            ndu[j][r] = right[r + 1] + left[j - r];
            float tmp = ndu[r][j - 1] * __builtin_amdgcn_rcpf(ndu[j][r]);
            ndu[r][j] = saved + right[r + 1] * tmp;
            saved = left[j - r] * tmp;
        }
        ndu[j][j] = saved;
    }
    N[0] = ndu[0][3]; N[1] = ndu[1][3]; N[2] = ndu[2][3]; N[3] = ndu[3][3];
    // First derivative closed form for p=3 (the reference's a[][] loop with k=1):
    // D[r] = 3 * ( (r>=1 ? ndu[r-1][2]/ndu[3][r-1] : 0) - (r<=2 ? ndu[r][2]/ndu[3][r] : 0) )
    float t0 = ndu[0][2] * __builtin_amdgcn_rcpf(ndu[3][0]);
    float t1 = ndu[1][2] * __builtin_amdgcn_rcpf(ndu[3][1]);
    float t2 = ndu[2][2] * __builtin_amdgcn_rcpf(ndu[3][2]);
    D[0] = -3.0f * t0;
    D[1] =  3.0f * (t0 - t1);
    D[2] =  3.0f * (t1 - t2);
    D[3] =  3.0f * t2;
}

__global__ __launch_bounds__(NTHREADS) void nurbs_surface_kernel(
    const float* __restrict__ ep, const float* __restrict__ cp,
    float* __restrict__ out, int n)
{
    __shared__ float s_cp4[NCP * NCP * 4];    // homogeneous control net, 16 KB
    __shared__ float s_stage[NCP * NCP * 3];  // raw (x,y,z) staging, 12 KB

    const int tid = threadIdx.x;

    // ---- Stage control net into LDS via CDNA5 async global->LDS copies ----
    // 12288 bytes = 768 x b128; 256 lanes x 3 ops. ASYNCcnt-tracked.
    {
        unsigned ldsBase = (unsigned)(unsigned long long)(const void*)&s_stage[0];
#pragma unroll
        for (int k = 0; k < 3; ++k) {
            unsigned off   = (unsigned)(tid + k * NTHREADS) * 16u;
            unsigned laddr = ldsBase + off;
            asm volatile("global_load_async_to_lds_b128 %0, %1, %2"
                         :: "v"(laddr), "v"(off), "s"(cp)
                         : "memory");
        }
        asm volatile("s_wait_asynccnt 0" ::: "memory");
        __syncthreads();
        // Expand (x,y,z) -> (x,y,z,1) in LDS
        for (int idx = tid; idx < NCP * NCP; idx += NTHREADS) {
            float x = s_stage[idx * 3 + 0];
            float y = s_stage[idx * 3 + 1];
            float z = s_stage[idx * 3 + 2];
            *reinterpret_cast<float4*>(&s_cp4[idx * 4]) = make_float4(x, y, z, 1.0f);
        }
        __syncthreads();
    }

    float* out_nrm = out + 4ull * (unsigned long long)n;
    const float2* ep2 = reinterpret_cast<const float2*>(ep);

    for (int i = blockIdx.x * NTHREADS + tid; i < n; i += gridDim.x * NTHREADS) {
        float2 uv = ep2[i];
        // span = floor(u * (n_cp - p)) + p ; kept as 0-based patch index here
        int se = (int)floorf(uv.x * 29.0f);
        int sn = (int)floorf(uv.y * 29.0f);
        se = se < 0 ? 0 : (se > 28 ? 28 : se);
        sn = sn < 0 ? 0 : (sn > 28 ? 28 : sn);

        float Ne[4], De[4], Nn[4], Dn[4];
        basis3(uv.x, se + 3, Ne, De);
        basis3(uv.y, sn + 3, Nn, Dn);

        float Sx = 0.f, Sy = 0.f, Sz = 0.f, Sw = 0.f;  // surface (homogeneous)
        float Ex = 0.f, Ey = 0.f, Ez = 0.f;            // d/du (xyz only)
        float Fx = 0.f, Fy = 0.f, Fz = 0.f;            // d/dv (xyz only)
#pragma unroll
        for (int r = 0; r < 4; ++r) {
            const float* row = &s_cp4[((se + r) * NCP + sn) * 4];
            const float ar = Ne[r], br = De[r];
#pragma unroll
            for (int s = 0; s < 4; ++s) {
                float4 g = *reinterpret_cast<const float4*>(&row[s * 4]);  // ds_load_b128
                float wS = ar * Nn[s];
                float wE = br * Nn[s];
                float wF = ar * Dn[s];
                Sx = fmaf(wS, g.x, Sx);
                Sy = fmaf(wS, g.y, Sy);
                Sz = fmaf(wS, g.z, Sz);
                Sw = fmaf(wS, g.w, Sw);
                Ex = fmaf(wE, g.x, Ex);
                Ey = fmaf(wE, g.y, Ey);
                Ez = fmaf(wE, g.z, Ez);
                Fx = fmaf(wF, g.x, Fx);
                Fy = fmaf(wF, g.y, Fy);
                Fz = fmaf(wF, g.z, Fz);
            }
        }

        float invw = __builtin_amdgcn_rcpf(Sw);
        // normal = normalize(cross(d_e.xyz, d_n.xyz))
        float cx = Ey * Fz - Ez * Fy;
        float cy = Ez * Fx - Ex * Fz;
        float cz = Ex * Fy - Ey * Fx;
        float il = __builtin_amdgcn_rsqf(fmaf(cx, cx, fmaf(cy, cy, cz * cz)));

        reinterpret_cast<float4*>(out)[i]     = make_float4(Sx * invw, Sy * invw, Sz * invw, 1.0f);
        reinterpret_cast<float4*>(out_nrm)[i] = make_float4(cx * il, cy * il, cz * il, 0.0f);
    }
}

extern "C" void kernel_launch(void* const* d_in, const int* in_sizes, int n_in,
                              void* d_out, int out_size, void* d_ws, size_t ws_size,
                              hipStream_t stream) {
    const float* ep = (const float*)d_in[0];   // (N_EVAL, 2) float32
    const float* cp = (const float*)d_in[1];   // (32, 32, 3) float32
    float* out = (float*)d_out;                // (N,4) points ++ (N,4) normals
    int n = in_sizes[0] / 2;
    if (n <= 0) return;
    int blocks = (n + 4 * NTHREADS - 1) / (4 * NTHREADS);  // ~4 points per thread
    if (blocks < 1) blocks = 1;
    if (blocks > 8192) blocks = 8192;
    hipLaunchKernelGGL(nurbs_surface_kernel, dim3(blocks), dim3(NTHREADS), 0, stream,
                       ep, cp, out, n);
}